// RoIAlign_28157805593218
// MI455X (gfx1250) — compile-verified
//
#include <hip/hip_runtime.h>
#include <hip/hip_bf16.h>
#include <stdint.h>

// Problem constants (match reference)
#define OUT_HW 7
#define Hf 200
#define Wf 304
#define Cf 256
#define HWf (Hf * Wf)             // 60800
#define PLANE ((size_t)Cf * HWf)  // elements per image, both layouts

typedef unsigned int v4u __attribute__((ext_vector_type(4)));
typedef int v4i __attribute__((ext_vector_type(4)));
typedef int v8i __attribute__((ext_vector_type(8)));

// ---------------------------------------------------------------------------
// CDNA5 split-counter waits with compile-time immediates. Inline asm +
// "memory" clobber so LDS reads cannot be hoisted above the wait.
// ---------------------------------------------------------------------------
template <int N>
__device__ __forceinline__ void wait_asynccnt() {
  asm volatile("s_wait_asynccnt %0" ::"n"(N) : "memory");
}
__device__ __forceinline__ void wait_tensorcnt0() {
  asm volatile("s_wait_tensorcnt %0" ::"n"(0) : "memory");
}

// ---------------------------------------------------------------------------
// Kernel 1: NCHW -> NHWC transpose (per image: [C, HW] -> [HW, C]).
// 32x32 tiles via LDS, fully coalesced both directions.
// ---------------------------------------------------------------------------
__global__ void nchw_to_nhwc(const float* __restrict__ in, float* __restrict__ out) {
  __shared__ float tile[32][33];
  const int n   = blockIdx.z;
  const int hw0 = blockIdx.x * 32;
  const int c0  = blockIdx.y * 32;
  const int tx  = threadIdx.x;   // 0..31
  const int ty  = threadIdx.y;   // 0..7

  const float* ip = in + ((size_t)n * Cf + c0) * HWf + hw0;
#pragma unroll
  for (int i = 0; i < 32; i += 8)
    tile[ty + i][tx] = ip[(size_t)(ty + i) * HWf + tx];

  __syncthreads();

  float* op = out + ((size_t)n * HWf + hw0) * Cf + c0;
#pragma unroll
  for (int i = 0; i < 32; i += 8)
    op[(size_t)(ty + i) * Cf + tx] = tile[tx][ty + i];
}

// ---------------------------------------------------------------------------
// ROI/tap math shared by all main kernels. Produces 16 tap plane-offsets
// (y*W+x, < 60800 so 16-bit safe) and 16 weights (pre-divided by 4 for the
// 2x2 sample mean; weight = 0 for invalid samples, matching the reference).
// ---------------------------------------------------------------------------
__device__ __forceinline__ void compute_taps(const float* __restrict__ rois, int r,
                                             int oy, int ox, int& bimg,
                                             int toff[16], float tw[16]) {
  const float* rp = rois + (size_t)r * 5;
  bimg = (int)rp[0];
  const float x1 = rp[1] * 0.25f - 0.5f;
  const float y1 = rp[2] * 0.25f - 0.5f;
  const float x2 = rp[3] * 0.25f - 0.5f;
  const float y2 = rp[4] * 0.25f - 0.5f;
  const float bw = (x2 - x1) * (1.0f / OUT_HW);
  const float bh = (y2 - y1) * (1.0f / OUT_HW);

#pragma unroll
  for (int s = 0; s < 4; ++s) {
    const int iy2 = s >> 1, ix2 = s & 1;
    const float y = y1 + oy * bh + ((float)iy2 + 0.5f) * bh * 0.5f;
    const float x = x1 + ox * bw + ((float)ix2 + 0.5f) * bw * 0.5f;
    const float valid =
        (y > -1.0f && y < (float)Hf && x > -1.0f && x < (float)Wf) ? 0.25f : 0.0f;
    const float yc = fminf(fmaxf(y, 0.0f), (float)(Hf - 1));
    const float xc = fminf(fmaxf(x, 0.0f), (float)(Wf - 1));
    const int y0 = (int)floorf(yc);
    const int x0 = (int)floorf(xc);
    const int y1i = min(y0 + 1, Hf - 1);
    const int x1i = min(x0 + 1, Wf - 1);
    const float ly = yc - (float)y0, lx = xc - (float)x0;
    const float hy = 1.0f - ly, hx = 1.0f - lx;

    toff[4 * s + 0] = y0 * Wf + x0;   tw[4 * s + 0] = hy * hx * valid;
    toff[4 * s + 1] = y0 * Wf + x1i;  tw[4 * s + 1] = hy * lx * valid;
    toff[4 * s + 2] = y1i * Wf + x0;  tw[4 * s + 2] = ly * hx * valid;
    toff[4 * s + 3] = y1i * Wf + x1i; tw[4 * s + 3] = ly * lx * valid;
  }
}

// Accumulate 8 channels/lane for tap T from the wave's LDS slab.
__device__ __forceinline__ void accum_one(const float* __restrict__ ldsp, int t,
                                          int lane, float w, float a[4], float b[4]) {
  const float4 v0 = *reinterpret_cast<const float4*>(ldsp + t * Cf + lane * 4);
  const float4 v1 = *reinterpret_cast<const float4*>(ldsp + t * Cf + 128 + lane * 4);
  a[0] = fmaf(w, v0.x, a[0]); a[1] = fmaf(w, v0.y, a[1]);
  a[2] = fmaf(w, v0.z, a[2]); a[3] = fmaf(w, v0.w, a[3]);
  b[0] = fmaf(w, v1.x, b[0]); b[1] = fmaf(w, v1.y, b[1]);
  b[2] = fmaf(w, v1.z, b[2]); b[3] = fmaf(w, v1.w, b[3]);
}

// Pipelined consumption for the async path: wait for the first 2*(T+1) async
// loads (they complete in issue order), then accumulate tap T.
template <int T>
__device__ __forceinline__ void accum_taps(const float* __restrict__ ldsp, int lane,
                                           const float tw[16], float a[4], float b[4]) {
  if constexpr (T < 16) {
    wait_asynccnt<30 - 2 * T>();
    accum_one(ldsp, T, lane, tw[T], a, b);
    accum_taps<T + 1>(ldsp, lane, tw, a, b);
  }
}

__device__ __forceinline__ void store_out(float* __restrict__ out, int r, int bin,
                                          int lane, const float a[4], const float b[4]) {
  // out layout: (R, C, 7, 7); lane owns channels 4*lane..+3 and 128+4*lane..+3
  float* op = out + (size_t)r * (Cf * 49) + bin;
#pragma unroll
  for (int k = 0; k < 4; ++k) {
    op[(size_t)(4 * lane + k) * 49] = a[k];
    op[(size_t)(128 + 4 * lane + k) * 49] = b[k];
  }
}

// ---------------------------------------------------------------------------
// Main kernel A (async vmem pipe): one wave per (roi, bin). 16 taps x 1KB
// gathered with global_load_async_to_lds_b128 (2 per tap), staged
// s_wait_asynccnt pipeline.
// ---------------------------------------------------------------------------
__global__ void __launch_bounds__(128) roi_align_nhwc_async(
    const float* __restrict__ nhwc, const float* __restrict__ rois,
    float* __restrict__ out, int job_base, int job_end) {
  __shared__ __align__(16) float lds_f[4 * 16 * Cf];  // 4 waves x 16KB = 64KB
  const int wave = threadIdx.x >> 5;
  const int lane = threadIdx.x & 31;
  const int job = job_base + blockIdx.x * 4 + wave;
  if (job >= job_end) return;  // wave-uniform

  const int r = job / 49;
  const int bin = job - r * 49;
  const int oy = bin / OUT_HW;
  const int ox = bin - oy * OUT_HW;

  int bimg, toff[16];
  float tw[16];
  compute_taps(rois, r, oy, ox, bimg, toff, tw);

  const float* img = nhwc + (size_t)bimg * PLANE;  // [HW][C], C contiguous
  float* ldsw_p = lds_f + wave * (16 * Cf);
  const unsigned lds_base = (unsigned)(uintptr_t)ldsw_p;  // low 32b flat = LDS offset

  // Issue all 32 async b128 loads (16 taps x 512B halves). ASYNCcnt <= 32 < 63.
#pragma unroll
  for (int t = 0; t < 16; ++t) {
    const unsigned l0 = lds_base + (unsigned)t * 1024u + (unsigned)lane * 16u;
    const unsigned long long g =
        (unsigned long long)(uintptr_t)(img + (size_t)toff[t] * Cf + lane * 4);
    asm volatile("global_load_async_to_lds_b128 %0, %1, off" ::"v"(l0), "v"(g)
                 : "memory");
    asm volatile("global_load_async_to_lds_b128 %0, %1, off offset:512" ::"v"(l0),
                 "v"(g)
                 : "memory");
  }

  float a[4] = {0.f, 0.f, 0.f, 0.f};
  float b[4] = {0.f, 0.f, 0.f, 0.f};
  accum_taps<0>(ldsw_p, lane, tw, a, b);
  store_out(out, r, bin, lane, a, b);
}

// ---------------------------------------------------------------------------
// Main kernel B (Tensor Data Mover, gather mode): one wave per (roi, bin).
// A single tensor_load_to_lds gathers all 16 tap rows (16-bit row indices,
// row width 256 fp32, tensor_dim0_stride = C) into the wave's 16KB LDS slab —
// identical layout to kernel A. Tracked with TENSORcnt.
// ---------------------------------------------------------------------------
__global__ void __launch_bounds__(128) roi_align_nhwc_tdm(
    const float* __restrict__ nhwc, const float* __restrict__ rois,
    float* __restrict__ out, int job_base, int job_end) {
  __shared__ __align__(16) float lds_f[4 * 16 * Cf];  // 4 waves x 16KB = 64KB
  const int wave = threadIdx.x >> 5;
  const int lane = threadIdx.x & 31;
  const int job = job_base + blockIdx.x * 4 + wave;
  if (job >= job_end) return;  // wave-uniform

  const int r = job / 49;
  const int bin = job - r * 49;
  const int oy = bin / OUT_HW;
  const int ox = bin - oy * OUT_HW;

  int bimg, toff[16];
  float tw[16];
  compute_taps(rois, r, oy, ox, bimg, toff, tw);

  const float* img = nhwc + (size_t)bimg * PLANE;
  float* ldsw_p = lds_f + wave * (16 * Cf);

#if __has_builtin(__builtin_amdgcn_tensor_load_to_lds)
  // ---- Build the Tensor DMA Descriptor (all fields wave-uniform) ----------
  const unsigned lds_addr =
      __builtin_amdgcn_readfirstlane((unsigned)(uintptr_t)ldsw_p);
  const unsigned long long gaddr = (unsigned long long)(uintptr_t)img;
  const unsigned ga_lo = __builtin_amdgcn_readfirstlane((unsigned)gaddr);
  const unsigned ga_hi =
      __builtin_amdgcn_readfirstlane((unsigned)(gaddr >> 32) & 0x01FFFFFFu);

  // Group 0: count=1 | gather_index_size=0(16b)<<30 | gather_mode=1<<31 ;
  //          lds_addr ; global_addr[31:0] ; global_addr[56:32] | type=2<<30
  v4u g0;
  g0[0] = 0x80000001u;
  g0[1] = lds_addr;
  g0[2] = ga_lo;
  g0[3] = ga_hi | 0x80000000u;

  // Group 1: wg_mask=0, data_size=2(4B)<<16 ; tensor_dim0=256@[79:48] ;
  //          tensor_dim1=60800@[111:80] ; tile_dim0=256@[127:112] ;
  //          tile_dim1(#indices)=16@[143:128] ; tensor_dim0_stride=256@[207:160]
  v8i g1;
  g1[0] = 0x00020000;                  // data_size = 4 bytes
  g1[1] = (int)(256u << 16);           // tensor_dim0[15:0] in [31:16]
  g1[2] = (int)(((unsigned)HWf & 0xFFFFu) << 16);  // tensor_dim1[15:0] in [31:16]
  g1[3] = (int)(256u << 16);           // tensor_dim1[31:16]=0, tile_dim0=256
  g1[4] = 16;                          // tile_dim1 = 16 valid row indices
  g1[5] = 256;                         // tensor_dim0_stride[31:0]
  g1[6] = 0;
  g1[7] = 0;

  // Groups 2/3: sixteen packed 16-bit row indices (y*W+x < 60800).
  v4i g2, g3;
#pragma unroll
  for (int k = 0; k < 4; ++k) {
    g2[k] = __builtin_amdgcn_readfirstlane((toff[2 * k + 1] << 16) |
                                           (toff[2 * k] & 0xFFFF));
    g3[k] = __builtin_amdgcn_readfirstlane((toff[8 + 2 * k + 1] << 16) |
                                           (toff[8 + 2 * k] & 0xFFFF));
  }

#if defined(__clang_major__) && (__clang_major__ >= 23)
  v8i gz = {0, 0, 0, 0, 0, 0, 0, 0};
  __builtin_amdgcn_tensor_load_to_lds(g0, g1, g2, g3, gz, 0);
#else
  __builtin_amdgcn_tensor_load_to_lds(g0, g1, g2, g3, 0);
#endif
  wait_tensorcnt0();
#else
  // Toolchain without the TDM builtin: fall back to the async-vmem gather.
#pragma unroll
  for (int t = 0; t < 16; ++t) {
    const unsigned l0 = (unsigned)(uintptr_t)ldsw_p + (unsigned)t * 1024u +
                        (unsigned)lane * 16u;
    const unsigned long long g =
        (unsigned long long)(uintptr_t)(img + (size_t)toff[t] * Cf + lane * 4);
    asm volatile("global_load_async_to_lds_b128 %0, %1, off" ::"v"(l0), "v"(g)
                 : "memory");
    asm volatile("global_load_async_to_lds_b128 %0, %1, off offset:512" ::"v"(l0),
                 "v"(g)
                 : "memory");
  }
  wait_asynccnt<0>();
#endif

  float a[4] = {0.f, 0.f, 0.f, 0.f};
  float b[4] = {0.f, 0.f, 0.f, 0.f};
#pragma unroll
  for (int t = 0; t < 16; ++t) accum_one(ldsw_p, t, lane, tw[t], a, b);
  store_out(out, r, bin, lane, a, b);
}

// ---------------------------------------------------------------------------
// Fallback: direct NCHW gather (used only if workspace can't hold NHWC copy).
// ---------------------------------------------------------------------------
__global__ void roi_align_nchw(const float* __restrict__ feat,
                               const float* __restrict__ rois,
                               float* __restrict__ out, int n_jobs) {
  const int wave = threadIdx.x >> 5;
  const int lane = threadIdx.x & 31;
  const int job = blockIdx.x * (blockDim.x >> 5) + wave;
  if (job >= n_jobs) return;

  const int r = job / 49;
  const int bin = job - r * 49;
  const int oy = bin / OUT_HW;
  const int ox = bin - oy * OUT_HW;

  int bimg, toff[16];
  float tw[16];
  compute_taps(rois, r, oy, ox, bimg, toff, tw);

  const float* base = feat + (size_t)bimg * PLANE;
  float acc[8] = {0.f, 0.f, 0.f, 0.f, 0.f, 0.f, 0.f, 0.f};
#pragma unroll
  for (int t = 0; t < 16; ++t) {
    const float w = tw[t];
    const float* p = base + toff[t];
#pragma unroll
    for (int i = 0; i < 8; ++i)
      acc[i] = fmaf(w, p[(size_t)(lane + i * 32) * HWf], acc[i]);
  }
  float* op = out + (size_t)r * (Cf * 49) + bin;
#pragma unroll
  for (int i = 0; i < 8; ++i) op[(size_t)(lane + i * 32) * 49] = acc[i];
}

// ---------------------------------------------------------------------------
extern "C" void kernel_launch(void* const* d_in, const int* in_sizes, int n_in,
                              void* d_out, int out_size, void* d_ws, size_t ws_size,
                              hipStream_t stream) {
  const float* features = (const float*)d_in[0];
  const float* rois = (const float*)d_in[1];
  float* out = (float*)d_out;

  const int n_img = in_sizes[0] / (int)PLANE;  // 2
  const int n_rois = in_sizes[1] / 5;          // 512
  const int n_jobs = n_rois * 49;              // 25088

  const size_t need = (size_t)n_img * PLANE * sizeof(float);  // 124.5 MB
  if (ws_size >= need) {
    float* nhwc = (float*)d_ws;
    dim3 tb(32, 8, 1);
    dim3 tg(HWf / 32, Cf / 32, n_img);
    nchw_to_nhwc<<<tg, tb, 0, stream>>>(features, nhwc);

    // Split jobs across the two independent data movers: the TDM engine
    // (gather-mode descriptors) and the async vector-memory pipe.
    const int n_tdm = n_jobs / 2;
    if (n_tdm > 0) {
      const int blocks = (n_tdm + 3) / 4;  // 4 waves (jobs) per block
      roi_align_nhwc_tdm<<<blocks, 128, 0, stream>>>(nhwc, rois, out, 0, n_tdm);
    }
    const int n_async = n_jobs - n_tdm;
    if (n_async > 0) {
      const int blocks = (n_async + 3) / 4;
      roi_align_nhwc_async<<<blocks, 128, 0, stream>>>(nhwc, rois, out, n_tdm,
                                                       n_jobs);
    }
  } else {
    const int blocks = (n_jobs + 7) / 8;  // 8 waves per block
    roi_align_nchw<<<blocks, 256, 0, stream>>>(features, rois, out, n_jobs);
  }
}